// LegalizeCLB_17016660426793
// MI455X (gfx1250) — compile-verified
//
#include <hip/hip_runtime.h>
#include <hip/hip_bf16.h>

// LegalizeCLB remaining-slice-sites kernel for MI455X (gfx1250, wave32).
//
// Pipeline:
//   P0: zero a 1 MB bitmap over the 2048*4096 site-id space (workspace).
//   P1: scatter assigned CLB sites into the bitmap; addr2site_map is sorted,
//       so we coalesce atomics per wave32 with a segmented suffix-OR
//       (ds_bpermute shuffles) -> ~16x fewer global_atomic_or_b32.
//   P2: slice_sites is ALSO sorted, so each 1024-site block touches a tiny
//       contiguous bitmap window (~45 words expected). Async-copy that window
//       into LDS (global_load_async_to_lds_b32 + s_wait_asynccnt, the CDNA5
//       ASYNCcnt path), then do all bit probes from LDS. Guarded fallback to
//       direct global probes if a window exceeds the 32 KB staging buffer.
// All touch-once streams use non-temporal hints; outputs are b128 NT stores.

typedef __attribute__((ext_vector_type(4))) int          v4i;
typedef __attribute__((ext_vector_type(4))) unsigned int v4u;
typedef __attribute__((ext_vector_type(4))) float        v4f;
typedef __attribute__((address_space(3)))  unsigned int  lds_u32;

#define NUM_SITES_X 2048
#define NUM_SITES_Y 4096
#define NUM_SITES   (NUM_SITES_X * NUM_SITES_Y)   // 8,388,608 site ids
#define MASK_WORDS  (NUM_SITES / 32)              // 262,144 u32 = 1 MB
#define GBLK        256                           // 8 wave32 per workgroup
#define GVEC        4
#define GTILE       (GBLK * GVEC)                 // 1024 sites per block
#define LDS_WORDS   8192                          // 32 KB staging window

// ---------------------------------------------------------------------------
// Pass 0: zero the bitmap (re-zeroed every call; harness doesn't re-poison).
// ---------------------------------------------------------------------------
__global__ void clb_zero_mask(unsigned int* __restrict__ mask) {
  int i = blockIdx.x * blockDim.x + threadIdx.x;        // MASK_WORDS/4 threads
  v4u z = {0u, 0u, 0u, 0u};
  ((v4u*)mask)[i] = z;
}

// ---------------------------------------------------------------------------
// Pass 1: scatter assigned CLB sites, wave32-coalesced atomics.
// ---------------------------------------------------------------------------
__global__ void clb_scatter_assigned(const int* __restrict__ sig,
                                     const int* __restrict__ addr,
                                     unsigned int* __restrict__ mask,
                                     int n) {
  int gid  = blockIdx.x * blockDim.x + threadIdx.x;
  int lane = threadIdx.x & 31;

  int          w    = -1;   // bitmap word index, -1 == inactive lane
  unsigned int bits = 0u;
  if (gid < n) {
    int s = __builtin_nontemporal_load(sig + gid);   // touch-once: TH=NT
    int a = __builtin_nontemporal_load(addr + gid);
    if (s > 0 && a >= 0 && a < NUM_SITES) {
      w    = a >> 5;
      bits = 1u << (a & 31);
    }
  }

  // Segmented suffix-OR over contiguous equal-word runs (all lanes execute).
#pragma unroll
  for (int off = 1; off < 32; off <<= 1) {
    int          ow = __shfl_down(w, off, 32);
    unsigned int ob = __shfl_down(bits, off, 32);
    if ((lane + off) < 32 && ow == w) bits |= ob;
  }
  int  up     = __shfl_up(w, 1, 32);
  bool leader = (lane == 0) || (up != w);

  if (w >= 0 && leader) {
    atomicOr(&mask[w], bits);                       // global_atomic_or_b32
  }
}

// ---------------------------------------------------------------------------
// Pass 2: LDS-staged gather (requires n % 4 == 0 for b128 alignment).
// ---------------------------------------------------------------------------
__global__ void clb_gather_staged_vec4(const int* __restrict__ sites,
                                       const float* __restrict__ scores,
                                       const unsigned int* __restrict__ mask,
                                       float* __restrict__ out_remaining,
                                       float* __restrict__ out_scores,
                                       int n) {
  __shared__ unsigned int lds_mask[LDS_WORDS];

  const int tid  = threadIdx.x;
  const int base = blockIdx.x * GTILE;
  const int last = min(base + GTILE, n) - 1;

  // Block-uniform bitmap window [w0, w1] (slice_sites is sorted).
  int s_lo = sites[base];
  int s_hi = sites[last];
  int c_lo = max(0, min(s_lo, NUM_SITES - 1));
  int c_hi = max(0, min(s_hi, NUM_SITES - 1));
  int w0 = min(c_lo, c_hi) >> 5;
  int w1 = max(c_lo, c_hi) >> 5;
  int nw = w1 - w0 + 1;
  const bool use_lds = (nw <= LDS_WORDS);

  if (use_lds) {
    // Async global->LDS copy of the window (GVS mode: SGPR base + VGPR off).
    unsigned long long gbase = (unsigned long long)(mask + w0);
    for (int i = tid; i < nw; i += GBLK) {
      unsigned int loff = (unsigned int)(unsigned long long)(lds_u32*)&lds_mask[i];
      unsigned int goff = (unsigned int)i * 4u;
      asm volatile("global_load_async_to_lds_b32 %0, %1, %2"
                   :: "v"(loff), "v"(goff), "s"(gbase) : "memory");
    }
    asm volatile("s_wait_asynccnt 0x0" ::: "memory");   // own ops complete
    __syncthreads();                                     // all waves' ops done
  }

  const int q = (base >> 2) + tid;                       // quad index
  if ((q << 2) >= n) return;                             // after the barrier

  v4i s  = __builtin_nontemporal_load((const v4i*)sites + q);
  v4f sc = __builtin_nontemporal_load((const v4f*)scores + q);

  v4f rem, av;
#pragma unroll
  for (int k = 0; k < 4; ++k) {
    int site = s[k];
    bool found = false;
    if (site >= 0 && site < NUM_SITES) {
      int idx = (site >> 5) - w0;
      unsigned int word;
      if (use_lds && idx >= 0 && idx < nw)
        word = lds_mask[idx];                            // ds_load_b32
      else
        word = mask[(unsigned)site >> 5];                // global fallback
      found = (word >> ((unsigned)site & 31u)) & 1u;
    }
    bool r = !found;
    rem[k] = r ? 1.0f : 0.0f;
    av[k]  = r ? sc[k] : 0.0f;
  }

  __builtin_nontemporal_store(rem, (v4f*)out_remaining + q);
  __builtin_nontemporal_store(av,  (v4f*)out_scores + q);
}

// Scalar fallback for n % 4 != 0 (keeps the +n output base alignment safe).
__global__ void clb_gather_scalar(const int* __restrict__ sites,
                                  const float* __restrict__ scores,
                                  const unsigned int* __restrict__ mask,
                                  float* __restrict__ out_remaining,
                                  float* __restrict__ out_scores,
                                  int n) {
  int j = blockIdx.x * blockDim.x + threadIdx.x;
  if (j >= n) return;
  int   site = __builtin_nontemporal_load(sites + j);
  float sc   = __builtin_nontemporal_load(scores + j);
  bool found = false;
  if (site >= 0 && site < NUM_SITES)
    found = (mask[(unsigned)site >> 5] >> ((unsigned)site & 31u)) & 1u;
  bool r = !found;
  __builtin_nontemporal_store(r ? 1.0f : 0.0f, out_remaining + j);
  __builtin_nontemporal_store(r ? sc : 0.0f,   out_scores + j);
}

extern "C" void kernel_launch(void* const* d_in, const int* in_sizes, int n_in,
                              void* d_out, int out_size, void* d_ws, size_t ws_size,
                              hipStream_t stream) {
  // setup_inputs() order:
  //   0: slice_sites      int32  [NUM_SLICE_SITES]   (sorted)
  //   1: site_det_sig_idx int32  [NUM_CLB_SITES]
  //   2: addr2site_map    int32  [NUM_CLB_SITES]     (sorted)
  //   3: site_scores      f32    [NUM_SLICE_SITES]
  const int*   slice_sites = (const int*)d_in[0];
  const int*   sig_idx     = (const int*)d_in[1];
  const int*   addr2site   = (const int*)d_in[2];
  const float* site_scores = (const float*)d_in[3];

  const int n_slice = in_sizes[0];
  const int n_clb   = in_sizes[1];

  float* out_remaining = (float*)d_out;            // first tuple element
  float* out_scores    = (float*)d_out + n_slice;  // second tuple element

  unsigned int* mask = (unsigned int*)d_ws;        // 1 MB bitmap in workspace
  (void)ws_size; (void)n_in; (void)out_size;

  // Pass 0: zero bitmap.
  clb_zero_mask<<<(MASK_WORDS / 4) / GBLK, GBLK, 0, stream>>>(mask);

  // Pass 1: scatter assigned sites.
  if (n_clb > 0) {
    int grid = (n_clb + GBLK - 1) / GBLK;
    clb_scatter_assigned<<<grid, GBLK, 0, stream>>>(sig_idx, addr2site, mask, n_clb);
  }

  // Pass 2: gather / membership test / masked select.
  if (n_slice > 0) {
    if ((n_slice & 3) == 0) {
      int grid = (n_slice + GTILE - 1) / GTILE;
      clb_gather_staged_vec4<<<grid, GBLK, 0, stream>>>(
          slice_sites, site_scores, mask, out_remaining, out_scores, n_slice);
    } else {
      int grid = (n_slice + GBLK - 1) / GBLK;
      clb_gather_scalar<<<grid, GBLK, 0, stream>>>(
          slice_sites, site_scores, mask, out_remaining, out_scores, n_slice);
    }
  }
}